// FreTransfer_33071248179686
// MI455X (gfx1250) — compile-verified
//
#include <hip/hip_runtime.h>

// Haar DWT (16,4,1024,1024) f32 -> (16,16,512,512) f32 on gfx1250.
// Per 16x16 tile:  Y = G^T * X * G  via chained V_WMMA_F32_16X16X4_F32,
// where G is the 16x16 banded Haar matrix (low taps cols 0-7, high cols 8-15).
// Quadrants of Y are the ll / lh / hl / hh 8x8 outputs.

typedef float v2f __attribute__((ext_vector_type(2)));
typedef float v8f __attribute__((ext_vector_type(8)));

// Full-EXEC exchange with lane ^ 16 (wave32): ds_swizzle group-of-32,
// xor_mask=0x10, or_mask=0, and_mask=0x1f  ->  offset 0x401F.
__device__ __forceinline__ float swap16(float v) {
    return __int_as_float(__builtin_amdgcn_ds_swizzle(__float_as_int(v), 0x401F));
}

__global__ __launch_bounds__(256) void haar_dwt16_wmma_kernel(
    const float* __restrict__ x,     // [16,4,1024,1024]
    const float* __restrict__ filt,  // [2,2] : {h00,h01,h10,h11}
    float* __restrict__ out)         // [16,16,512,512]  (ll,hl,lh,hh over C)
{
    constexpr int W  = 1024;
    constexpr int Ho = 512, Wo = 512;

    const int lane = threadIdx.x & 31;
    const int wave = threadIdx.x >> 5;
    const int task = blockIdx.x * 8 + wave;   // 65536 wave-tasks total

    const int img = task >> 10;               // b*4 + c   (0..63)
    const int th  = (task >> 4) & 63;         // 16-row tile band
    const int twg = task & 15;                // group of 4 tiles along W

    const int b = img >> 2;
    const int c = img & 3;

    const float h00 = filt[0], h01 = filt[1], h10 = filt[2], h11 = filt[3];

    const int  half = lane >> 4;   // 0 = lanes 0-15, 1 = lanes 16-31
    const int  l15  = lane & 15;
    const bool hi   = (half != 0);

    // Unified constant fragment set. For chunk k, vgpr v, lane (l15, half):
    //   as B-operand it holds G[4k + 2*half + v][l15]
    //   as A-operand it holds G^T[l15][4k + 2*half + v]   (same value).
    // G[r][j] != 0  iff  (r>>1) == (j & 7); value = taps(j>=8, r&1).
    // With r = 4k + 2*half + v:  (r>>1) == 2k + half, independent of v.
    const float tap0 = (l15 >= 8) ? h10 : h00;   // r even tap
    const float tap1 = (l15 >= 8) ? h11 : h01;   // r odd  tap
    v2f F[4];
    #pragma unroll
    for (int k = 0; k < 4; ++k) {
        const bool nz = ((2 * k + half) == (l15 & 7));
        F[k][0] = nz ? tap0 : 0.0f;
        F[k][1] = nz ? tap1 : 0.0f;
    }

    // A-operand loads for X: lane l15 = matrix row; lane-half selects K pair.
    const float* xrow = x + (size_t)img * (1024u * 1024u)
                          + (size_t)(th * 16 + l15) * W
                          + twg * 64 + 2 * half;

    // Output plane per lane-octet: 0-7 -> ll(0), 8-15 -> lh(2),
    // 16-23 -> hl(1), 24-31 -> hh(3).
    const int p = ((lane >> 3) & 1) * 2 + half;
    float* orow = out + (size_t)b * (16u * Ho * Wo)
                      + (size_t)(p * 4 + c) * (size_t)(Ho * Wo)
                      + (size_t)(th * 8) * Wo
                      + twg * 32 + (lane & 7);

    #pragma unroll
    for (int i = 0; i < 4; ++i) {            // 4 adjacent tiles: 2 full cachelines/row
        const float* xt = xrow + i * 16;
        v2f x0 = *(const v2f*)(xt +  0);
        v2f x1 = *(const v2f*)(xt +  4);
        v2f x2 = *(const v2f*)(xt +  8);
        v2f x3 = *(const v2f*)(xt + 12);

        // T = X * G   (row decomposition), K=16 as 4 chained K=4 WMMAs.
        v8f t = {0,0,0,0,0,0,0,0};
        t = __builtin_amdgcn_wmma_f32_16x16x4_f32(false, x0, false, F[0], (short)0, t, false, false);
        t = __builtin_amdgcn_wmma_f32_16x16x4_f32(false, x1, false, F[1], (short)0, t, false, false);
        t = __builtin_amdgcn_wmma_f32_16x16x4_f32(false, x2, false, F[2], (short)0, t, false, false);
        t = __builtin_amdgcn_wmma_f32_16x16x4_f32(false, x3, false, F[3], (short)0, t, false, false);

        // D-layout -> B-operand layout. Row r of T sits in vgpr (r&7),
        // lane-half (r>=8); chunk k needs row 4k+2*half+v. Mismatched halves
        // come from lane^16. All swizzles run UNCONDITIONALLY at full EXEC
        // (required: disabled source lanes would return 0), then cndmask.
        const float sw0 = swap16(t[0]);
        const float sw1 = swap16(t[1]);
        const float sw2 = swap16(t[2]);
        const float sw3 = swap16(t[3]);
        const float sw4 = swap16(t[4]);
        const float sw5 = swap16(t[5]);
        const float sw6 = swap16(t[6]);
        const float sw7 = swap16(t[7]);

        v2f B0, B1, B2, B3;
        B0[0] = hi ? sw2 : t[0];   B0[1] = hi ? sw3 : t[1];   // rows 0..3
        B1[0] = hi ? sw6 : t[4];   B1[1] = hi ? sw7 : t[5];   // rows 4..7
        B2[0] = hi ? t[2] : sw0;   B2[1] = hi ? t[3] : sw1;   // rows 8..11
        B3[0] = hi ? t[6] : sw4;   B3[1] = hi ? t[7] : sw5;   // rows 12..15

        // Y = G^T * T  (column decomposition); A-operand is the same F set.
        v8f y = {0,0,0,0,0,0,0,0};
        y = __builtin_amdgcn_wmma_f32_16x16x4_f32(false, F[0], false, B0, (short)0, y, false, false);
        y = __builtin_amdgcn_wmma_f32_16x16x4_f32(false, F[1], false, B1, (short)0, y, false, false);
        y = __builtin_amdgcn_wmma_f32_16x16x4_f32(false, F[2], false, B2, (short)0, y, false, false);
        y = __builtin_amdgcn_wmma_f32_16x16x4_f32(false, F[3], false, B3, (short)0, y, false, false);

        // Y quadrants -> ll/hl/lh/hh planes; 32B contiguous per lane-octet.
        #pragma unroll
        for (int r = 0; r < 8; ++r)
            orow[r * Wo + i * 8] = y[r];
    }
}

extern "C" void kernel_launch(void* const* d_in, const int* in_sizes, int n_in,
                              void* d_out, int out_size, void* d_ws, size_t ws_size,
                              hipStream_t stream) {
    const float* x    = (const float*)d_in[0];
    const float* filt = (const float*)d_in[1];
    float*       out  = (float*)d_out;

    // 64 images * 64 tile-rows * 16 tile-groups = 65536 wave-tasks,
    // 8 waves per 256-thread block -> 8192 blocks, exact (EXEC always full).
    haar_dwt16_wmma_kernel<<<dim3(8192), dim3(256), 0, stream>>>(x, filt, out);
}